// ntm_read_head_5488968204591
// MI455X (gfx1250) — compile-verified
//
#include <hip/hip_runtime.h>
#include <math.h>

#define B_ 512
#define C_ 256
#define N_ 2048
#define D_ 64
#define EMIT 70
#define PSTRIDE 80

typedef __attribute__((ext_vector_type(2))) float v2f;
typedef __attribute__((ext_vector_type(8))) float v8f;

__device__ __forceinline__ float softplus_f(float x) {
    return (x > 20.f) ? x : log1pf(expf(x));
}

// ---------------------------------------------------------------------------
// Kernel 1: fc = tanh(co @ W + b); derive head params per batch.
// params[b*80 + ...]: [0..63]=k, 64=beta, 65=g, 66..68=s, 69=gamma, 70=|k|
// ---------------------------------------------------------------------------
__global__ __launch_bounds__(128) void ntm_fc_kernel(
    const float* __restrict__ co, const float* __restrict__ W,
    const float* __restrict__ bias, float* __restrict__ params)
{
    __shared__ float sco[C_];
    __shared__ float sfc[EMIT];
    const int b = blockIdx.x;
    const int t = threadIdx.x;
    for (int i = t; i < C_; i += 128) sco[i] = co[(size_t)b * C_ + i];
    __syncthreads();
    if (t < EMIT) {
        float acc = bias[t];
        #pragma unroll 8
        for (int c = 0; c < C_; ++c) acc = fmaf(sco[c], W[c * EMIT + t], acc);
        sfc[t] = tanhf(acc);
    }
    __syncthreads();
    float* pb = params + (size_t)b * PSTRIDE;
    if (t < 64) {
        pb[t] = sfc[t];
    } else if (t == 64) {
        pb[64] = softplus_f(sfc[64]);                     // beta
    } else if (t == 65) {
        pb[65] = 1.f / (1.f + expf(-sfc[65]));            // g
    } else if (t == 66) {                                 // softmax over 3 shift logits
        float m = fmaxf(sfc[66], fmaxf(sfc[67], sfc[68]));
        float e0 = expf(sfc[66] - m), e1 = expf(sfc[67] - m), e2 = expf(sfc[68] - m);
        float s = e0 + e1 + e2;
        pb[66] = e0 / s; pb[67] = e1 / s; pb[68] = e2 / s;
    } else if (t == 67) {
        pb[69] = 1.f + softplus_f(sfc[69]);               // gamma
    } else if (t == 68) {
        float s = 0.f;
        #pragma unroll 8
        for (int d = 0; d < 64; ++d) s += sfc[d] * sfc[d];
        pb[70] = sqrtf(s);                                // |k|
    }
}

// ---------------------------------------------------------------------------
// Block reductions (512 threads = 16 waves, wave32)
// ---------------------------------------------------------------------------
__device__ __forceinline__ float block_max(float v, float* red, int tid, int lane, int wv) {
    #pragma unroll
    for (int off = 16; off > 0; off >>= 1) v = fmaxf(v, __shfl_down(v, off));
    if (lane == 0) red[wv] = v;
    __syncthreads();
    if (tid == 0) {
        float m = red[0];
        for (int w = 1; w < 16; ++w) m = fmaxf(m, red[w]);
        red[16] = m;
    }
    __syncthreads();
    float r = red[16];
    __syncthreads();
    return r;
}

__device__ __forceinline__ float block_sum(float v, float* red, int tid, int lane, int wv) {
    #pragma unroll
    for (int off = 16; off > 0; off >>= 1) v += __shfl_down(v, off);
    if (lane == 0) red[wv] = v;
    __syncthreads();
    if (tid == 0) {
        float s = red[0];
        for (int w = 1; w < 16; ++w) s += red[w];
        red[16] = s;
    }
    __syncthreads();
    float r = red[16];
    __syncthreads();
    return r;
}

// ---------------------------------------------------------------------------
// Kernel 2: one block per batch. WMMA content-dot, softmax addressing chain,
// WMMA readout. 512 threads = 16 waves; each wave owns 128 rows of M[b].
// ---------------------------------------------------------------------------
__global__ __launch_bounds__(512) void ntm_addr_read_kernel(
    const float* __restrict__ M, const float* __restrict__ wprev,
    const float* __restrict__ params, float* __restrict__ out_r,
    float* __restrict__ out_w)
{
    __shared__ float dotbuf[N_];   // dot -> w_g -> r_partials
    __shared__ float magbuf[N_];   // |m|^2 -> w_t
    __shared__ float pblk[72];
    __shared__ float red[17];

    const int b    = blockIdx.x;
    const int tid  = threadIdx.x;
    const int lane = tid & 31;
    const int wv   = tid >> 5;
    const int half = lane >> 4;    // 0: lanes 0-15 (K=0,1)   1: lanes 16-31 (K=2,3)
    const int lr   = lane & 15;
    const float* Mb = M + (size_t)b * N_ * D_;

    if (tid < 71) pblk[tid] = params[(size_t)b * PSTRIDE + tid];
    __syncthreads();

    // ---- Phase A: dot[b,n] = k . M[b,n,:] and |M[b,n,:]|^2, WMMA f32 16x16x4
    v2f kb[16];                    // B-frags: k broadcast into all 16 columns
    #pragma unroll
    for (int c = 0; c < 16; ++c) {
        const int ko = 4 * c + 2 * half;
        kb[c].x = pblk[ko]; kb[c].y = pblk[ko + 1];
    }
    for (int t = 0; t < 8; ++t) {
        const int n0 = wv * 128 + t * 16;
        const float* rowp = Mb + (size_t)(n0 + lr) * D_ + 2 * half;
        v8f acc = {};
        float magp = 0.f;
        #pragma unroll
        for (int c = 0; c < 16; ++c) {
            v2f a = *(const v2f*)(rowp + 4 * c);       // A-frag: M[n0+lr, 4c+2h .. +1]
            magp = fmaf(a.x, a.x, magp);
            magp = fmaf(a.y, a.y, magp);
            acc = __builtin_amdgcn_wmma_f32_16x16x4_f32(
                false, a, false, kb[c], (short)0, acc, false, false);
        }
        const float magfull = magp + __shfl_xor(magp, 16);
        if (lane < 16) magbuf[n0 + lr] = magfull;
        if (lane == 0) {
            #pragma unroll
            for (int r = 0; r < 8; ++r) dotbuf[n0 + r] = acc[r];       // M=r, N=0
        } else if (lane == 16) {
            #pragma unroll
            for (int r = 0; r < 8; ++r) dotbuf[n0 + 8 + r] = acc[r];   // M=8+r, N=0
        }
    }
    __syncthreads();

    // ---- Phase B: softmax(beta*cos), interpolate, shift, sharpen, renorm
    const float beta = pblk[64], g = pblk[65];
    const float s0 = pblk[66], s1 = pblk[67], s2 = pblk[68];
    const float gamma = pblk[69], kmag = pblk[70];

    float z[4];
    #pragma unroll
    for (int j = 0; j < 4; ++j) {
        const int n = tid + 512 * j;
        z[j] = beta * dotbuf[n] / (kmag * sqrtf(magbuf[n]));
    }
    float vmax = fmaxf(fmaxf(z[0], z[1]), fmaxf(z[2], z[3]));
    const float rmax = block_max(vmax, red, tid, lane, wv);

    float er[4], lsum = 0.f;
    #pragma unroll
    for (int j = 0; j < 4; ++j) { er[j] = expf(z[j] - rmax); lsum += er[j]; }
    const float rsum = block_sum(lsum, red, tid, lane, wv);

    #pragma unroll
    for (int j = 0; j < 4; ++j) {                      // w_g = g*w_c + (1-g)*w_prev
        const int n = tid + 512 * j;
        const float wc = er[j] / rsum;
        dotbuf[n] = g * wc + (1.f - g) * wprev[(size_t)b * N_ + n];
    }
    __syncthreads();

    float pw[4], psum = 0.f;
    #pragma unroll
    for (int j = 0; j < 4; ++j) {                      // circular conv + sharpen
        const int n = tid + 512 * j;
        const float wt = s0 * dotbuf[(n + 1) & (N_ - 1)]
                       + s1 * dotbuf[n]
                       + s2 * dotbuf[(n - 1) & (N_ - 1)];
        pw[j] = powf(wt, gamma);
        psum += pw[j];
    }
    const float tsum = block_sum(psum, red, tid, lane, wv);

    #pragma unroll
    for (int j = 0; j < 4; ++j) {
        const int n = tid + 512 * j;
        const float wn = pw[j] / tsum;
        magbuf[n] = wn;                                // keep w_t for readout
        out_w[(size_t)b * N_ + n] = wn;
    }
    __syncthreads();

    // ---- Phase C: r[b,:] = w_t @ M[b], WMMA (A = w broadcast rows, B = M tile)
    v8f racc[4];
    #pragma unroll
    for (int t = 0; t < 4; ++t) racc[t] = (v8f){};
    const int nbase = wv * 128;
    for (int c = 0; c < 32; ++c) {
        const int nc = nbase + 4 * c;
        v2f av;                                        // A-frag: w[nc + 2h .. +1]
        av.x = magbuf[nc + 2 * half];
        av.y = magbuf[nc + 2 * half + 1];
        #pragma unroll
        for (int t = 0; t < 4; ++t) {
            const float* p = Mb + (size_t)(nc + 2 * half) * D_ + t * 16 + lr;
            v2f bv;                                    // B-frag: M[nc+K, d0+lane]
            bv.x = p[0];
            bv.y = p[D_];
            racc[t] = __builtin_amdgcn_wmma_f32_16x16x4_f32(
                false, av, false, bv, (short)0, racc[t], false, false);
        }
    }
    if (lane < 16) {                                   // row M=0 = partial r[d0+lane]
        #pragma unroll
        for (int t = 0; t < 4; ++t) dotbuf[wv * 64 + t * 16 + lane] = racc[t][0];
    }
    __syncthreads();
    if (tid < 64) {
        float s = 0.f;
        #pragma unroll
        for (int w = 0; w < 16; ++w) s += dotbuf[w * 64 + tid];
        out_r[(size_t)b * D_ + tid] = s;
    }
}

// ---------------------------------------------------------------------------
extern "C" void kernel_launch(void* const* d_in, const int* in_sizes, int n_in,
                              void* d_out, int out_size, void* d_ws, size_t ws_size,
                              hipStream_t stream) {
    const float* co    = (const float*)d_in[0];   // (512,256)
    const float* wprev = (const float*)d_in[1];   // (512,2048)
    const float* M     = (const float*)d_in[2];   // (512,2048,64)
    const float* W     = (const float*)d_in[3];   // (256,70)
    const float* bias  = (const float*)d_in[4];   // (70,)

    float* out_r = (float*)d_out;                 // (512,64)
    float* out_w = (float*)d_out + B_ * D_;       // (512,2048)
    float* params = (float*)d_ws;                 // 512*80 floats

    ntm_fc_kernel<<<B_, 128, 0, stream>>>(co, W, bias, params);
    ntm_addr_read_kernel<<<B_, 512, 0, stream>>>(M, wprev, params, out_r, out_w);
}